// AttnDecoder_61881888800872
// MI455X (gfx1250) — compile-verified
//
#include <hip/hip_runtime.h>
#include <hip/hip_bf16.h>

typedef float v2f __attribute__((ext_vector_type(2)));
typedef float v8f __attribute__((ext_vector_type(8)));

#define B_ 32
#define S_ 4096
#define H_ 512
#define ENC_ 1024
#define V_ 32000
#define G3_ 1536   // 3*H
#define CAT_ 1536  // IN + ENC

// ---------------- WMMA helper: D = A(16x4 f32) * B(4x16 f32) + C ----------------
__device__ __forceinline__ v8f wmma_k4(v2f a, v2f b, v8f c) {
  return __builtin_amdgcn_wmma_f32_16x16x4_f32(
      /*neg_a=*/false, a, /*neg_b=*/false, b,
      /*c_mod=*/(short)0, c, /*reuse_a=*/false, /*reuse_b=*/false);
}

// out[32,N] = A[32,K] @ W[N,K]^T + bias[N]   (one wave -> 32x16 tile)
__global__ __launch_bounds__(256) void gemm32_wT(
    const float* __restrict__ A, const float* __restrict__ W,
    const float* __restrict__ bias, float* __restrict__ out, int N, int K) {
  int wave = (blockIdx.x * blockDim.x + threadIdx.x) >> 5;
  int lane = threadIdx.x & 31;
  int n0 = wave * 16;
  if (n0 >= N) return;
  int ln = lane & 15;
  int kh = (lane >> 4) << 1;  // 0 or 2
  float bv = bias ? bias[n0 + ln] : 0.0f;
  v8f acc0, acc1;
#pragma unroll
  for (int i = 0; i < 8; ++i) { acc0[i] = bv; acc1[i] = bv; }
  const float* wr = W + (size_t)(n0 + ln) * K + kh;
  const float* a0 = A + (size_t)ln * K + kh;
  const float* a1 = A + (size_t)(16 + ln) * K + kh;
  for (int k = 0; k < K; k += 4) {
    v2f bf = *reinterpret_cast<const v2f*>(wr + k);
    v2f af0 = *reinterpret_cast<const v2f*>(a0 + k);
    v2f af1 = *reinterpret_cast<const v2f*>(a1 + k);
    acc0 = wmma_k4(af0, bf, acc0);
    acc1 = wmma_k4(af1, bf, acc1);
  }
  int rb = (lane >> 4) << 3;  // 0 or 8
#pragma unroll
  for (int r = 0; r < 8; ++r) {
    out[(size_t)(r + rb) * N + n0 + ln] = acc0[r];
    out[(size_t)(16 + r + rb) * N + n0 + ln] = acc1[r];
  }
}

// out[32,N] = A[32,K] @ Bm[K,N] + bias[N]   (row-major B, used for q = h1 @ attn_W)
__global__ __launch_bounds__(256) void gemm32_b(
    const float* __restrict__ A, const float* __restrict__ Bm,
    const float* __restrict__ bias, float* __restrict__ out, int N, int K) {
  int wave = (blockIdx.x * blockDim.x + threadIdx.x) >> 5;
  int lane = threadIdx.x & 31;
  int n0 = wave * 16;
  if (n0 >= N) return;
  int ln = lane & 15;
  int kh = (lane >> 4) << 1;
  float bv = bias ? bias[n0 + ln] : 0.0f;
  v8f acc0, acc1;
#pragma unroll
  for (int i = 0; i < 8; ++i) { acc0[i] = bv; acc1[i] = bv; }
  const float* a0 = A + (size_t)ln * K + kh;
  const float* a1 = A + (size_t)(16 + ln) * K + kh;
  for (int k = 0; k < K; k += 4) {
    v2f bf;
    bf.x = Bm[(size_t)(k + kh) * N + n0 + ln];
    bf.y = Bm[(size_t)(k + kh + 1) * N + n0 + ln];
    v2f af0 = *reinterpret_cast<const v2f*>(a0 + k);
    v2f af1 = *reinterpret_cast<const v2f*>(a1 + k);
    acc0 = wmma_k4(af0, bf, acc0);
    acc1 = wmma_k4(af1, bf, acc1);
  }
  int rb = (lane >> 4) << 3;
#pragma unroll
  for (int r = 0; r < 8; ++r) {
    out[(size_t)(r + rb) * N + n0 + ln] = acc0[r];
    out[(size_t)(16 + r + rb) * N + n0 + ln] = acc1[r];
  }
}

// cat0[b, 0:512] = x[b]; cat0[b, 512:1536] = context[b]
__global__ void build_cat0(const float* __restrict__ x, const float* __restrict__ ctx,
                           float* __restrict__ cat) {
  int i = blockIdx.x * 256 + threadIdx.x;
  if (i >= B_ * CAT_) return;
  int b = i / CAT_, j = i - b * CAT_;
  cat[i] = (j < 512) ? x[b * 512 + j] : ctx[b * ENC_ + (j - 512)];
}

// GRU gate combine (PyTorch order r,z,n); optionally mirror h into d_out and cat buffer
__global__ void gru_combine(const float* __restrict__ gi, const float* __restrict__ gh,
                            const float* __restrict__ hprev, float* __restrict__ h,
                            float* __restrict__ h_out, float* __restrict__ cat_dst) {
  int i = blockIdx.x * 256 + threadIdx.x;
  if (i >= B_ * H_) return;
  int b = i >> 9, j = i & (H_ - 1);
  const float* gib = gi + (size_t)b * G3_;
  const float* ghb = gh + (size_t)b * G3_;
  float r = 1.0f / (1.0f + __expf(-(gib[j] + ghb[j])));
  float z = 1.0f / (1.0f + __expf(-(gib[H_ + j] + ghb[H_ + j])));
  float n = tanhf(gib[2 * H_ + j] + r * ghb[2 * H_ + j]);
  float hv = (1.0f - z) * n + z * hprev[i];
  h[i] = hv;
  if (h_out) h_out[i] = hv;
  if (cat_dst) cat_dst[(size_t)b * G3_ + j] = hv;
}

__global__ void bias_dot_kernel(const float* __restrict__ h1,
                                const float* __restrict__ attn_b, float* __restrict__ bd) {
  int b = threadIdx.x;
  if (b >= B_) return;
  float a = 0.0f;
  for (int h = 0; h < H_; ++h) a += h1[b * H_ + h] * attn_b[h];
  bd[b] = a;
}

__device__ __forceinline__ float wred_sum(float v) {
#pragma unroll
  for (int m = 16; m >= 1; m >>= 1) v += __shfl_xor(v, m, 32);
  return v;
}
__device__ __forceinline__ float wred_max(float v) {
#pragma unroll
  for (int m = 16; m >= 1; m >>= 1) v = fmaxf(v, __shfl_xor(v, m, 32));
  return v;
}

// energies[b,s] = q[b] . enc[b,s] + bias_dot[b]   (one wave per row, float4 streams)
__global__ __launch_bounds__(256) void energies_kernel(
    const float* __restrict__ q, const float* __restrict__ enc,
    const float* __restrict__ bd, float* __restrict__ energ) {
  int row = (blockIdx.x * blockDim.x + threadIdx.x) >> 5;
  int lane = threadIdx.x & 31;
  if (row >= B_ * S_) return;
  int b = row >> 12;
  const float4* er = reinterpret_cast<const float4*>(enc + (size_t)row * ENC_);
  const float4* qr = reinterpret_cast<const float4*>(q + (size_t)b * ENC_);
  float acc = 0.0f;
#pragma unroll
  for (int k = 0; k < 8; ++k) {
    float4 e = er[lane + k * 32];
    float4 w = qr[lane + k * 32];
    acc += e.x * w.x + e.y * w.y + e.z * w.z + e.w * w.w;
  }
  acc = wred_sum(acc);
  if (lane == 0) energ[row] = acc + bd[b];
}

// row softmax over S=4096, block(1024) per batch element
__global__ __launch_bounds__(1024) void softmax_kernel(const float* __restrict__ e,
                                                       float* __restrict__ attn) {
  __shared__ float red[32];
  __shared__ float bval;
  int b = blockIdx.x, t = threadIdx.x, lane = t & 31, wid = t >> 5;
  float4 v = reinterpret_cast<const float4*>(e + (size_t)b * S_)[t];
  float m = fmaxf(fmaxf(v.x, v.y), fmaxf(v.z, v.w));
  m = wred_max(m);
  if (lane == 0) red[wid] = m;
  __syncthreads();
  if (t < 32) {
    float mm = wred_max(red[lane]);
    if (lane == 0) bval = mm;
  }
  __syncthreads();
  m = bval;
  float4 ev;
  ev.x = __expf(v.x - m); ev.y = __expf(v.y - m);
  ev.z = __expf(v.z - m); ev.w = __expf(v.w - m);
  float s = ev.x + ev.y + ev.z + ev.w;
  s = wred_sum(s);
  __syncthreads();
  if (lane == 0) red[wid] = s;
  __syncthreads();
  if (t < 32) {
    float ss = wred_sum(red[lane]);
    if (lane == 0) bval = ss;
  }
  __syncthreads();
  float inv = 1.0f / bval;
  ev.x *= inv; ev.y *= inv; ev.z *= inv; ev.w *= inv;
  reinterpret_cast<float4*>(attn + (size_t)b * S_)[t] = ev;
}

// new_context[b,f] = sum_s attn[b,s]*enc[b,s,f]; also fill cat_out[b, 512+f]
__global__ __launch_bounds__(256) void context_kernel(
    const float* __restrict__ attn, const float* __restrict__ enc,
    float* __restrict__ ctx_out, float* __restrict__ cat_out) {
  __shared__ float aw[S_];
  int b = blockIdx.x >> 2;
  int f = ((blockIdx.x & 3) << 8) + threadIdx.x;
  const float* ab = attn + (size_t)b * S_;
  for (int s = threadIdx.x; s < S_; s += 256) aw[s] = ab[s];
  __syncthreads();
  const float* eb = enc + (size_t)b * S_ * ENC_ + f;
  float acc = 0.0f;
#pragma unroll 4
  for (int s = 0; s < S_; ++s) acc += aw[s] * eb[(size_t)s * ENC_];
  ctx_out[(size_t)b * ENC_ + f] = acc;
  cat_out[(size_t)b * G3_ + 512 + f] = acc;
}

extern "C" void kernel_launch(void* const* d_in, const int* in_sizes, int n_in,
                              void* d_out, int out_size, void* d_ws, size_t ws_size,
                              hipStream_t stream) {
  const float* x      = (const float*)d_in[0];
  const float* ctx    = (const float*)d_in[1];
  const float* hidden = (const float*)d_in[2];   // [2,32,512]
  const float* enc    = (const float*)d_in[3];   // [32,4096,1024]
  const float* attn_W = (const float*)d_in[4];   // [512,1024]
  const float* attn_b = (const float*)d_in[5];
  const float* W_ih0  = (const float*)d_in[6];
  const float* W_hh0  = (const float*)d_in[7];
  const float* b_ih0  = (const float*)d_in[8];
  const float* b_hh0  = (const float*)d_in[9];
  const float* W_ih1  = (const float*)d_in[10];
  const float* W_hh1  = (const float*)d_in[11];
  const float* b_ih1  = (const float*)d_in[12];
  const float* b_hh1  = (const float*)d_in[13];
  const float* out_W  = (const float*)d_in[14];  // [32000,1536]
  const float* out_b  = (const float*)d_in[15];
  float* out = (float*)d_out;

  // d_out layout: output[32,32000] | new_context[32,1024] | hidden[2,32,512] | attn[32,4096]
  float* o_logits = out;
  float* o_ctx    = out + (size_t)B_ * V_;
  float* o_hid    = o_ctx + (size_t)B_ * ENC_;
  float* o_attn   = o_hid + 2 * (size_t)B_ * H_;

  // workspace layout (floats)
  float* w = (float*)d_ws;
  float* cat0   = w;                 // 32*1536
  float* gi0    = cat0 + B_ * CAT_;  // 32*1536
  float* gh0    = gi0 + B_ * G3_;
  float* h0     = gh0 + B_ * G3_;    // 32*512
  float* gi1    = h0 + B_ * H_;
  float* gh1    = gi1 + B_ * G3_;
  float* h1     = gh1 + B_ * G3_;    // 32*512
  float* q      = h1 + B_ * H_;      // 32*1024
  float* bd     = q + B_ * ENC_;     // 32
  float* energ  = bd + 32;           // 32*4096
  float* catout = energ + B_ * S_;   // 32*1536

  // 1) concat input+context
  build_cat0<<<(B_ * CAT_ + 255) / 256, 256, 0, stream>>>(x, ctx, cat0);
  // 2) GRU layer 0 gates (WMMA)
  gemm32_wT<<<G3_ / 16 / 8, 256, 0, stream>>>(cat0, W_ih0, b_ih0, gi0, G3_, CAT_);
  gemm32_wT<<<G3_ / 16 / 8, 256, 0, stream>>>(hidden, W_hh0, b_hh0, gh0, G3_, H_);
  gru_combine<<<(B_ * H_ + 255) / 256, 256, 0, stream>>>(gi0, gh0, hidden, h0, o_hid, nullptr);
  // 3) GRU layer 1 gates (WMMA)
  gemm32_wT<<<G3_ / 16 / 8, 256, 0, stream>>>(h0, W_ih1, b_ih1, gi1, G3_, H_);
  gemm32_wT<<<G3_ / 16 / 8, 256, 0, stream>>>(hidden + B_ * H_, W_hh1, b_hh1, gh1, G3_, H_);
  gru_combine<<<(B_ * H_ + 255) / 256, 256, 0, stream>>>(gi1, gh1, hidden + B_ * H_, h1,
                                                         o_hid + B_ * H_, catout);
  // 4) q = h1 @ attn_W  (reassociated attention, avoids 137-GFLOP tmp)
  gemm32_b<<<ENC_ / 16 / 8, 256, 0, stream>>>(h1, attn_W, nullptr, q, ENC_, H_);
  bias_dot_kernel<<<1, 32, 0, stream>>>(h1, attn_b, bd);
  // 5) energies: one streaming pass over encoder_outputs (536 MB)
  energies_kernel<<<(B_ * S_) / 8, 256, 0, stream>>>(q, enc, bd, energ);
  // 6) softmax -> attn weights (written straight to d_out)
  softmax_kernel<<<B_, 1024, 0, stream>>>(energ, o_attn);
  // 7) context: second pass over encoder_outputs
  context_kernel<<<B_ * 4, 256, 0, stream>>>(o_attn, enc, o_ctx, catout);
  // 8) vocab projection [32,1536] @ [1536->32000] (WMMA, streams out_W once)
  gemm32_wT<<<V_ / 16 / 8, 256, 0, stream>>>(catout, out_W, out_b, o_logits, V_, G3_);
}